// DADMMLASSO_17368847745615
// MI455X (gfx1250) — compile-verified
//
#include <hip/hip_runtime.h>

// ---------------- problem constants ----------------
#define PN      5            // nodes
#define MROWS   500          // rows of A
#define NCOLS   2000         // cols of A
#define BB      1024         // batch
#define KITERS  10           // ADMM iterations
#define MP      100          // rows per node
#define AXLD    112          // padded leading dim for AX buffer (7*16)
#define JT5     5            // n-tiles per wave column group (125 = 25*5 exact)
#define NGRP    80           // JT5*16 columns per block panel
#define KLD     102          // padded k-stride for transposed panel (8B pairs, banking)
#define KCH     100          // K chunk for ax staging (2000 = 20*100 exact)
#define MPAD    112          // padded rows for ax B panel (7*16)
#define ROWSPAN 512          // second row-tile offset: 32 tiles * 16 rows

typedef __attribute__((ext_vector_type(2))) float v2f;
typedef __attribute__((ext_vector_type(8))) float v8f;

// D = A(16x4) * B(4x16) + C  -> V_WMMA_F32_16X16X4_F32
__device__ __forceinline__ v8f wmma4(v2f a, v2f b, v8f c) {
    return __builtin_amdgcn_wmma_f32_16x16x4_f32(
        false, a, false, b, (short)0, c, false, false);
}

// ---------------- kernels ----------------

__global__ void zero_kernel(float4* __restrict__ dst, long long n4) {
    long long i = (long long)blockIdx.x * blockDim.x + threadIdx.x;
    if (i < n4) dst[i] = make_float4(0.f, 0.f, 0.f, 0.f);
}

// Atb[p,b,n] = sum_m b[b, p*MP+m] * A[p*MP+m, n]      (K = MP = 100)
// grid (25, 4, PN), block 256 (8 waves). Wave w -> row-tiles it0, it0+32.
// B panel staged TRANSPOSED: sBt[n][k] so each fragment is one ds_load_b64.
__global__ void __launch_bounds__(256, 1)
atb_kernel(const float* __restrict__ A, const float* __restrict__ b,
           float* __restrict__ Atb) {
    const int p    = blockIdx.z;
    const int n0   = blockIdx.x * NGRP;
    const int it0  = blockIdx.y * 8 + (threadIdx.x >> 5);   // 0..31
    const int r0   = it0 * 16;
    const int r1   = r0 + ROWSPAN;
    const float* Ap = A + (size_t)p * MP * NCOLS;

    __shared__ float sBt[NGRP][KLD];   // [80][102]: pair (k,k+1) contiguous
    for (int idx = threadIdx.x; idx < MP * NGRP / 4; idx += 256) {
        const int kk = idx / (NGRP / 4);
        const int c4 = (idx % (NGRP / 4)) * 4;
        float4 v = *(const float4*)&Ap[(size_t)kk * NCOLS + n0 + c4];
        sBt[c4 + 0][kk] = v.x;
        sBt[c4 + 1][kk] = v.y;
        sBt[c4 + 2][kk] = v.z;
        sBt[c4 + 3][kk] = v.w;
    }
    __syncthreads();

    const int lane = threadIdx.x & 31;
    const int nl   = lane & 15;
    const int kh   = (lane >> 4) << 1;

    // strength-reduced row pointers (A-fragments) and LDS base (B-fragments)
    const float* pa0 = b + (size_t)(r0 + nl) * MROWS + p * MP + kh;
    const float* pa1 = b + (size_t)(r1 + nl) * MROWS + p * MP + kh;
    const float* ps  = &sBt[nl][kh];

    v8f acc0[JT5] = {};
    v8f acc1[JT5] = {};
    for (int k0 = 0; k0 < MP; k0 += 4) {
        v2f af0; af0.x = pa0[0]; af0.y = pa0[1]; pa0 += 4;
        v2f af1; af1.x = pa1[0]; af1.y = pa1[1]; pa1 += 4;
#pragma unroll
        for (int j = 0; j < JT5; ++j) {
            v2f bf = *(const v2f*)(ps + j * 16 * KLD);  // one ds_load_b64
            acc0[j] = wmma4(af0, bf, acc0[j]);
            acc1[j] = wmma4(af1, bf, acc1[j]);
        }
        ps += 4;
    }

    float* out = Atb + (size_t)p * BB * NCOLS;
    const int ih = (lane >> 4) * 8;
#pragma unroll
    for (int j = 0; j < JT5; ++j) {
        const int col = n0 + j * 16 + nl;
#pragma unroll
        for (int v = 0; v < 8; ++v) {
            out[(size_t)(r0 + ih + v) * NCOLS + col] = acc0[j][v];
            out[(size_t)(r1 + ih + v) * NCOLS + col] = acc1[j][v];
        }
    }
}

// AX[b,m] = sum_n X[p,b,n] * A_p[m,n]                  (K = NCOLS = 2000)
// grid (4), block 256. Wave w -> row-tiles it0, it0+32; all 7 m-tiles.
// B panel A_p[0:112(pad), kc:kc+100] staged per K-chunk (44.8 KB),
// already k-contiguous -> fragments are single ds_load_b64.
__global__ void __launch_bounds__(256, 1)
ax_kernel(const float* __restrict__ A, const float* __restrict__ X,
          float* __restrict__ axbuf, int p) {
    const int it0 = blockIdx.x * 8 + (threadIdx.x >> 5);    // 0..31
    const int r0  = it0 * 16;
    const int r1  = r0 + ROWSPAN;
    const float* Ap = A + (size_t)p * MP * NCOLS;
    const float* Xp = X + (size_t)p * BB * NCOLS;

    __shared__ float sB[MPAD][KCH];

    const int lane = threadIdx.x & 31;
    const int ml   = lane & 15;
    const int kh   = (lane >> 4) << 1;

    // X rows are contiguous in k across all chunks: just keep incrementing.
    const float* px0 = Xp + (size_t)(r0 + ml) * NCOLS + kh;
    const float* px1 = Xp + (size_t)(r1 + ml) * NCOLS + kh;

    v8f acc0[7] = {};
    v8f acc1[7] = {};
    for (int kc = 0; kc < NCOLS; kc += KCH) {
        __syncthreads();
        for (int idx = threadIdx.x; idx < MPAD * KCH / 4; idx += 256) {
            const int m  = idx / (KCH / 4);
            const int c4 = (idx % (KCH / 4)) * 4;
            float4 v = make_float4(0.f, 0.f, 0.f, 0.f);
            if (m < MP) {
                v = *(const float4*)&Ap[(size_t)m * NCOLS + kc + c4];
                if (kc + KCH < NCOLS)
                    __builtin_prefetch(&Ap[(size_t)m * NCOLS + kc + KCH + c4], 0, 1);
            }
            *(float4*)&sB[m][c4] = v;
        }
        __syncthreads();

        for (int k0 = 0; k0 < KCH; k0 += 4) {
            v2f af0; af0.x = px0[0]; af0.y = px0[1]; px0 += 4;
            v2f af1; af1.x = px1[0]; af1.y = px1[1]; px1 += 4;
            const float* ps = &sB[ml][k0 + kh];
#pragma unroll
            for (int j = 0; j < 7; ++j) {
                v2f bf = *(const v2f*)(ps + j * 16 * KCH);  // 8B-aligned
                acc0[j] = wmma4(af0, bf, acc0[j]);
                acc1[j] = wmma4(af1, bf, acc1[j]);
            }
        }
    }

    const int ih = (lane >> 4) * 8;
#pragma unroll
    for (int j = 0; j < 7; ++j) {
        const int col = j * 16 + ml;   // m
        if (col < MP) {
#pragma unroll
            for (int v = 0; v < 8; ++v) {
                axbuf[(size_t)(r0 + ih + v) * AXLD + col] = acc0[j][v];
                axbuf[(size_t)(r1 + ih + v) * AXLD + col] = acc1[j][v];
            }
        }
    }
}

// AtAX + fused primal update:
// X[p] = X[p] - a1*(AtAX - Atb + 2*a0*X + a2*sign(X) + 2*U - a0*(X[l]+X[r]))
// grid (25, 4), block 256. Same transposed panel staging as atb_kernel.
__global__ void __launch_bounds__(256, 1)
xupd_kernel(const float* __restrict__ A, const float* __restrict__ axbuf,
            const float* __restrict__ Atb, const float* __restrict__ hyp,
            const int* __restrict__ nbr,
            float* __restrict__ X, const float* __restrict__ U,
            int p, int k) {
    const int n0  = blockIdx.x * NGRP;
    const int it0 = blockIdx.y * 8 + (threadIdx.x >> 5);    // 0..31
    const int r0  = it0 * 16;
    const int r1  = r0 + ROWSPAN;
    const float* Ap = A + (size_t)p * MP * NCOLS;

    __shared__ float sBt[NGRP][KLD];   // [80][102]
    for (int idx = threadIdx.x; idx < MP * NGRP / 4; idx += 256) {
        const int kk = idx / (NGRP / 4);
        const int c4 = (idx % (NGRP / 4)) * 4;
        float4 v = *(const float4*)&Ap[(size_t)kk * NCOLS + n0 + c4];
        sBt[c4 + 0][kk] = v.x;
        sBt[c4 + 1][kk] = v.y;
        sBt[c4 + 2][kk] = v.z;
        sBt[c4 + 3][kk] = v.w;
    }
    __syncthreads();

    const int lane = threadIdx.x & 31;
    const int nl   = lane & 15;
    const int kh   = (lane >> 4) << 1;

    const float* pa0 = axbuf + (size_t)(r0 + nl) * AXLD + kh;
    const float* pa1 = axbuf + (size_t)(r1 + nl) * AXLD + kh;
    const float* ps  = &sBt[nl][kh];

    v8f acc0[JT5] = {};
    v8f acc1[JT5] = {};
    for (int k0 = 0; k0 < MP; k0 += 4) {          // K = 100 = 25*4 exactly
        v2f af0; af0.x = pa0[0]; af0.y = pa0[1]; pa0 += 4;
        v2f af1; af1.x = pa1[0]; af1.y = pa1[1]; pa1 += 4;
#pragma unroll
        for (int j = 0; j < JT5; ++j) {
            v2f bf = *(const v2f*)(ps + j * 16 * KLD);  // one ds_load_b64
            acc0[j] = wmma4(af0, bf, acc0[j]);
            acc1[j] = wmma4(af1, bf, acc1[j]);
        }
        ps += 4;
    }

    const float a0 = fabsf(hyp[(k * PN + p) * 4 + 0]);
    const float a1 = fabsf(hyp[(k * PN + p) * 4 + 1]);
    const float a2 = fabsf(hyp[(k * PN + p) * 4 + 2]);
    const int pl = nbr[p * 2 + 0];
    const int pr = nbr[p * 2 + 1];
    const size_t stride = (size_t)BB * NCOLS;
    float*       Xp = X + (size_t)p * stride;
    const float* Xl = X + (size_t)pl * stride;
    const float* Xr = X + (size_t)pr * stride;
    const float* Up = U + (size_t)p * stride;
    const float* Tp = Atb + (size_t)p * stride;

    const int ih = (lane >> 4) * 8;
#pragma unroll
    for (int j = 0; j < JT5; ++j) {
        const int col = n0 + j * 16 + nl;
#pragma unroll
        for (int v = 0; v < 8; ++v) {
            {
                const size_t off = (size_t)(r0 + ih + v) * NCOLS + col;
                const float xo  = Xp[off];
                const float sgn = (xo > 0.f) ? 1.f : ((xo < 0.f) ? -1.f : 0.f);
                const float g   = acc0[j][v] - Tp[off] + 2.f * a0 * xo + a2 * sgn
                                  + 2.f * Up[off] - a0 * (Xl[off] + Xr[off]);
                Xp[off] = xo - a1 * g;
            }
            {
                const size_t off = (size_t)(r1 + ih + v) * NCOLS + col;
                const float xo  = Xp[off];
                const float sgn = (xo > 0.f) ? 1.f : ((xo < 0.f) ? -1.f : 0.f);
                const float g   = acc1[j][v] - Tp[off] + 2.f * a0 * xo + a2 * sgn
                                  + 2.f * Up[off] - a0 * (Xl[off] + Xr[off]);
                Xp[off] = xo - a1 * g;
            }
        }
    }
}

// U[p] += a3 * (2*X[p] - X[l] - X[r])   (float4 vectorized)
__global__ void dual_kernel(float* __restrict__ XU, const float* __restrict__ hyp,
                            const int* __restrict__ nbr, int k) {
    const size_t stride4 = (size_t)BB * NCOLS / 4;
    const size_t total4  = (size_t)PN * stride4;
    size_t i = (size_t)blockIdx.x * blockDim.x + threadIdx.x;
    if (i >= total4) return;
    const int p = (int)(i / stride4);
    const size_t off = i - (size_t)p * stride4;
    const float a3 = fabsf(hyp[(k * PN + p) * 4 + 3]);
    const int pl = nbr[p * 2 + 0];
    const int pr = nbr[p * 2 + 1];
    const float4* X = (const float4*)XU;
    float4*       Uv = (float4*)(XU + (size_t)PN * BB * NCOLS);
    const float4 x  = X[i];
    const float4 xl = X[(size_t)pl * stride4 + off];
    const float4 xr = X[(size_t)pr * stride4 + off];
    float4 u = Uv[i];
    u.x += a3 * (2.f * x.x - xl.x - xr.x);
    u.y += a3 * (2.f * x.y - xl.y - xr.y);
    u.z += a3 * (2.f * x.z - xl.z - xr.z);
    u.w += a3 * (2.f * x.w - xl.w - xr.w);
    Uv[i] = u;
}

// ---------------- host entry ----------------
extern "C" void kernel_launch(void* const* d_in, const int* in_sizes, int n_in,
                              void* d_out, int out_size, void* d_ws, size_t ws_size,
                              hipStream_t stream) {
    (void)in_sizes; (void)n_in; (void)out_size; (void)ws_size;
    const float* A   = (const float*)d_in[0];   // [500,2000]
    const float* b   = (const float*)d_in[1];   // [1024,500]
    const float* hyp = (const float*)d_in[2];   // [10,5,4]
    const int*   nbr = (const int*)d_in[3];     // [5,2]

    float* X = (float*)d_out;                              // [5,1024,2000]
    float* U = X + (size_t)PN * BB * NCOLS;                // [5,1024,2000]
    float* Atb   = (float*)d_ws;                           // [5,1024,2000]
    float* axbuf = Atb + (size_t)PN * BB * NCOLS;          // [1024, AXLD]

    // X = U = 0
    {
        long long n4 = (2LL * PN * BB * NCOLS) / 4;
        int blocks = (int)((n4 + 255) / 256);
        zero_kernel<<<blocks, 256, 0, stream>>>((float4*)d_out, n4);
    }
    // Atb = b_p @ A_p (all nodes)
    atb_kernel<<<dim3(25, 4, PN), 256, 0, stream>>>(A, b, Atb);

    const int colors[3][2] = {{0, 2}, {1, 3}, {4, -1}};
    for (int k = 0; k < KITERS; ++k) {
        for (int c = 0; c < 3; ++c) {
            for (int s = 0; s < 2; ++s) {
                const int p = colors[c][s];
                if (p < 0) continue;
                ax_kernel  <<<4,           256, 0, stream>>>(A, X, axbuf, p);
                xupd_kernel<<<dim3(25, 4), 256, 0, stream>>>(A, axbuf, Atb, hyp, nbr,
                                                             X, U, p, k);
            }
        }
        {
            long long n4 = ((long long)PN * BB * NCOLS) / 4;
            int blocks = (int)((n4 + 255) / 256);
            dual_kernel<<<blocks, 256, 0, stream>>>(X, hyp, nbr, k);
        }
    }
}